// LegacySeqFirstSelfAttention_grad_7679401525959
// MI455X (gfx1250) — compile-verified
//
#include <hip/hip_runtime.h>
#include <hip/hip_bf16.h>

// ---------------------------------------------------------------------------
// LegacySeqFirstSelfAttention forward on MI455X (gfx1250), f32 WMMA path.
// S=2048, B=2, H=1024, nh=16, hph=64. Outputs: context | scores | probs (f32).
// Bandwidth-bound: ~1.09 GB mandatory f32 output writes + 537 MB probs re-read
// (~70 us floor @ 23.3 TB/s). Each wave computes a 16x64 strip (4 WMMA tiles
// sharing one A fragment) so probs is read from HBM exactly once in the
// context GEMM. All >L2-sized streams use nontemporal hints so the reused
// tensors (qkv 50 MB, v slices, mask 17 MB) stay resident in the 192 MB L2.
// ---------------------------------------------------------------------------

typedef float v2f __attribute__((ext_vector_type(2)));
typedef float v8f __attribute__((ext_vector_type(8)));

#define S_LEN 2048
#define BATCH 2
#define HID   1024
#define NH    16
#define HPH   64
#define H3    3072           // 3*HID
#define GRP   192            // 3*HPH: per-head q|k|v group inside the qkv row
#define NORM  0.125f         // rsqrt(64)
#define BH    (BATCH * NH)   // 32
#define WAVES 8              // 256-thread block = 8 wave32
#define NT    4              // N-tiles per wave (16x64 strip per wave)

// D = A(16x4) * B(4x16) + C, f32. EXEC must be all-ones (guaranteed: full
// wave32 blocks, wave-uniform control flow only).
__device__ __forceinline__ v8f wmma4(v2f a, v2f b, v8f c) {
  return __builtin_amdgcn_wmma_f32_16x16x4_f32(false, a, false, b, (short)0, c,
                                               false, false);
}

// ---------------------------------------------------------------------------
// Kernel 1: qkv[(s*B+b)*3H + n] = X[(s*B+b)] . W[n] + bias[n]
// M = S*B = 4096, N = 3H = 3072, K = 1024. 16x64 strip per wave: A fragment
// (X) loaded once per k-step, reused for 4 B (W) fragments.
// ---------------------------------------------------------------------------
__global__ void __launch_bounds__(256)
qkv_gemm_kernel(const float* __restrict__ X, const float* __restrict__ W,
                const float* __restrict__ bias, float* __restrict__ qkv) {
  const int lane = threadIdx.x & 31;
  const int wave = threadIdx.x >> 5;
  const int tile = blockIdx.x * WAVES + wave;   // 256 * 48 strips
  const int tn_strips = H3 / (16 * NT);         // 48
  const int tm = tile / tn_strips;
  const int tn = tile % tn_strips;

  const int half = lane >> 4;
  const int l15  = lane & 15;
  const int row  = tm * 16 + l15;               // A-matrix row (m)
  const int col0 = tn * (16 * NT) + l15;        // first B col (n)

  const float* __restrict__ ap = X + (size_t)row * HID + 2 * half;
  const float* __restrict__ bp[NT];
#pragma unroll
  for (int j = 0; j < NT; ++j)
    bp[j] = W + (size_t)(col0 + 16 * j) * HID + 2 * half;

  v8f c[NT] = {};
#pragma unroll 4
  for (int k0 = 0; k0 < HID; k0 += 4) {
    v2f a = *(const v2f*)(ap + k0);
#pragma unroll
    for (int j = 0; j < NT; ++j) {
      v2f b = *(const v2f*)(bp[j] + k0);
      c[j] = wmma4(a, b, c[j]);
    }
  }

#pragma unroll
  for (int j = 0; j < NT; ++j) {
    const int col = col0 + 16 * j;
    const float bn = bias[col];
#pragma unroll
    for (int r = 0; r < 8; ++r) {
      const int m = tm * 16 + r + 8 * half;     // C/D layout: M = r + 8*half
      qkv[(size_t)m * H3 + col] = c[j][r] + bn;
    }
  }
}

// ---------------------------------------------------------------------------
// Kernel 2: scores[bh,s,t] = (q[s,b,h,:] . k[t,b,h,:]) * NORM
//           probs [bh,s,t] = exp(scores + mask[s,t])
// K = hph = 64 -> 16 k-steps. 16x64 strip per wave: q fragment reused for 4
// k-tiles. Both 537 MB output streams are written nontemporal: scores is
// never re-read, and probs is consumed bh-major by kernel 3 long after its
// L2 lines would have been evicted — NT keeps the reused qkv in L2 instead.
// ---------------------------------------------------------------------------
__global__ void __launch_bounds__(256)
attn_scores_kernel(const float* __restrict__ qkv, const float* __restrict__ mask,
                   float* __restrict__ scores, float* __restrict__ probs) {
  const int lane = threadIdx.x & 31;
  const int wave = threadIdx.x >> 5;
  const int tile = blockIdx.x * WAVES + wave;   // 32 * 128 * 32 strips
  const int TSTRIP = S_LEN / (16 * NT);         // 32 t-strips
  const int bh  = tile / ((S_LEN / 16) * TSTRIP);
  const int rem = tile % ((S_LEN / 16) * TSTRIP);
  const int ts  = rem / TSTRIP;
  const int ttm = rem % TSTRIP;
  const int b = bh >> 4;
  const int h = bh & 15;

  const int half = lane >> 4;
  const int l15  = lane & 15;
  const int srow  = ts * 16 + l15;              // query position (A rows)
  const int tcol0 = ttm * (16 * NT) + l15;      // first key position (B cols)

  const float* __restrict__ qp =
      qkv + ((size_t)srow * BATCH + b) * H3 + h * GRP + 2 * half;
  const float* __restrict__ kp[NT];
#pragma unroll
  for (int j = 0; j < NT; ++j)
    kp[j] = qkv + ((size_t)(tcol0 + 16 * j) * BATCH + b) * H3 + h * GRP + HPH +
            2 * half;

  v8f c[NT] = {};
#pragma unroll
  for (int k0 = 0; k0 < HPH; k0 += 4) {
    v2f a = *(const v2f*)(qp + k0);
#pragma unroll
    for (int j = 0; j < NT; ++j) {
      v2f bb = *(const v2f*)(kp[j] + k0);
      c[j] = wmma4(a, bb, c[j]);
    }
  }

  const size_t base = (size_t)bh * S_LEN * S_LEN;
#pragma unroll
  for (int j = 0; j < NT; ++j) {
    const int tcol = tcol0 + 16 * j;
#pragma unroll
    for (int r = 0; r < 8; ++r) {
      const int sr = ts * 16 + r + 8 * half;
      const float sc = c[j][r] * NORM;
      const size_t idx = base + (size_t)sr * S_LEN + tcol;
      __builtin_nontemporal_store(sc, scores + idx);      // TH=NT stream-out
      __builtin_nontemporal_store(
          __expf(sc + mask[(size_t)sr * S_LEN + tcol]), probs + idx);
    }
  }
}

// ---------------------------------------------------------------------------
// Kernel 3: ctx[s,b,h,d] = sum_t probs[bh,s,t] * v[t,b,h,d]
// K = S = 2048 -> 512 k-steps. One wave computes the full 16(s) x 64(d) head
// strip: the probs A-fragment (537 MB, HBM-resident) is loaded once with a
// nontemporal hint (pure stream; don't evict the reused v slices from L2);
// v (L2-resident, 16 MB) supplies the B fragments.
// ---------------------------------------------------------------------------
__global__ void __launch_bounds__(256)
attn_context_kernel(const float* __restrict__ probs, const float* __restrict__ qkv,
                    float* __restrict__ ctx) {
  const int lane = threadIdx.x & 31;
  const int wave = threadIdx.x >> 5;
  const int tile = blockIdx.x * WAVES + wave;   // 32 * 128 strips
  const int ts = tile & 127;                    // 128 s-tiles
  const int bh = tile >> 7;                     // 32 batch-heads
  const int b = bh >> 4;
  const int h = bh & 15;

  const int half = lane >> 4;
  const int l15  = lane & 15;
  const int srow = ts * 16 + l15;

  const float* __restrict__ ap =
      probs + (size_t)bh * S_LEN * S_LEN + (size_t)srow * S_LEN + 2 * half;
  const float* __restrict__ vbase =
      qkv + (size_t)b * H3 + h * GRP + 2 * HPH + l15;     // + t*(B*3H) + 16*j

  v8f c[NT] = {};
#pragma unroll 2
  for (int k0 = 0; k0 < S_LEN; k0 += 4) {
    v2f a = __builtin_nontemporal_load((const v2f*)(ap + k0));  // TH=NT stream
    const int t0 = k0 + 2 * half;
    const float* __restrict__ vp0 = vbase + (size_t)t0 * (BATCH * H3);
#pragma unroll
    for (int j = 0; j < NT; ++j) {
      v2f bb;
      bb.x = vp0[16 * j];
      bb.y = vp0[BATCH * H3 + 16 * j];
      c[j] = wmma4(a, bb, c[j]);
    }
  }

#pragma unroll
  for (int j = 0; j < NT; ++j) {
    const int dcol = 16 * j + l15;
#pragma unroll
    for (int r = 0; r < 8; ++r) {
      const int s = ts * 16 + r + 8 * half;
      ctx[(size_t)s * (BATCH * HID) + (size_t)b * HID + h * HPH + dcol] = c[j][r];
    }
  }
}

// ---------------------------------------------------------------------------
extern "C" void kernel_launch(void* const* d_in, const int* in_sizes, int n_in,
                              void* d_out, int out_size, void* d_ws, size_t ws_size,
                              hipStream_t stream) {
  const float* hidden = (const float*)d_in[0];   // [S, B, H]
  const float* mask   = (const float*)d_in[1];   // [1, S, S]
  const float* W      = (const float*)d_in[2];   // [3H, H]
  const float* bias   = (const float*)d_in[3];   // [3H]

  float* out    = (float*)d_out;
  float* ctx    = out;                                          // S*B*H
  float* scores = out + (size_t)S_LEN * BATCH * HID;            // BH*S*S
  float* probs  = scores + (size_t)BH * S_LEN * S_LEN;          // BH*S*S

  float* qkv = (float*)d_ws;  // [S, B, 3H] = 50.3 MB scratch

  // 1) QKV projection: 256 x 48 strips of 16x64
  {
    const int strips = (S_LEN * BATCH / 16) * (H3 / (16 * NT)); // 12288
    qkv_gemm_kernel<<<strips / WAVES, 256, 0, stream>>>(hidden, W, bias, qkv);
  }
  // 2) scores + exp: 32 * 128 * 32 strips of 16x64
  {
    const int strips = BH * (S_LEN / 16) * (S_LEN / (16 * NT)); // 131072
    attn_scores_kernel<<<strips / WAVES, 256, 0, stream>>>(qkv, mask, scores, probs);
  }
  // 3) context: 32 * 128 strips of 16x64
  {
    const int strips = BH * (S_LEN / 16);                       // 4096
    attn_context_kernel<<<strips / WAVES, 256, 0, stream>>>(probs, qkv, ctx);
  }
}